// HierSoftmaxNLL_3702261809386
// MI455X (gfx1250) — compile-verified
//
#include <hip/hip_runtime.h>

typedef __attribute__((ext_vector_type(2))) float v2f;
typedef __attribute__((ext_vector_type(8))) float v8f;

#define NSCORES 9999   // N_NODES - 1
#define NGROUPS 625    // internal nodes 0..624; group id == parent node id
#define BATCH   1024

// One wave32 handles one group (16 children) x 16 batch rows.
// Lane layout (matches CDNA5 f32 WMMA B-matrix 4x16 layout):
//   n = lane & 15  -> matrix column  = batch row b0+n
//   khalf = lane>>4 -> which K-half the lane carries per chunk
// For chunk c (K = 4c..4c+3): lane holds K = 4c+2*khalf and 4c+2*khalf+1.
// With A = ones(16x4), D[m][n] = sum_k B[k][n] regardless of K<->slot mapping,
// and every lane's acc[0] = column sum for column (lane&15).
__global__ __launch_bounds__(256) void lse_wmma_kernel(const float* __restrict__ scores,
                                                       float* __restrict__ lse) {
    int g = blockIdx.x * blockDim.y + threadIdx.y;   // wave-uniform
    if (g >= NGROUPS) return;                        // uniform exit: EXEC stays full
    int b0   = blockIdx.y * 16;
    int lane = threadIdx.x;
    int n     = lane & 15;
    int khalf = lane >> 4;
    int row   = b0 + n;
    const float* srow = scores + (long)row * NSCORES;

    float v[8];
#pragma unroll
    for (int c = 0; c < 4; ++c) {
        int k0   = 4 * c + 2 * khalf;
        int col0 = g * 16 + k0;
        v[2*c]     = (col0     < NSCORES) ? srow[col0]     : -__builtin_inff();
        v[2*c + 1] = (col0 + 1 < NSCORES) ? srow[col0 + 1] : -__builtin_inff();
    }

    // column (batch-row) max: 8-local max, then combine K-halves across lane^16
    float m = v[0];
#pragma unroll
    for (int j = 1; j < 8; ++j) m = fmaxf(m, v[j]);
    m = fmaxf(m, __shfl_xor(m, 16, 32));

    float e[8];
#pragma unroll
    for (int j = 0; j < 8; ++j) e[j] = __expf(v[j] - m);  // exp(-inf)=0 for the pad slot

    v2f ones; ones.x = 1.0f; ones.y = 1.0f;
    v8f acc = {0.f, 0.f, 0.f, 0.f, 0.f, 0.f, 0.f, 0.f};
#pragma unroll
    for (int c = 0; c < 4; ++c) {
        v2f bmat; bmat.x = e[2*c]; bmat.y = e[2*c + 1];
        // D = ones(16x4) x E(4x16) + C : column sums of exp tile, full f32
        acc = __builtin_amdgcn_wmma_f32_16x16x4_f32(false, ones, false, bmat,
                                                    (short)0, acc, false, false);
    }

    float lseval = m + __logf(acc[0]);
    if (khalf == 0) lse[(long)row * NGROUPS + g] = lseval;
}

// One block of 1024 threads: each walks its leaf->root path (<=4 levels),
// sums score - lse per level, tree-reduces, writes mean NLL.
__global__ __launch_bounds__(1024) void path_nll_kernel(const float* __restrict__ scores,
                                                        const int* __restrict__ labels,
                                                        const int* __restrict__ label_order,
                                                        const float* __restrict__ lse,
                                                        float* __restrict__ out) {
    __shared__ float red[BATCH];
    int b = threadIdx.x;
    int node = label_order[labels[b]];   // leaf node id
    float acc = 0.f;
    while (node != 0) {
        int p = node - 1;                // score / flat index
        int gg = p >> 4;                 // parent node id == group id
        acc += scores[(long)b * NSCORES + p] - lse[(long)b * NGROUPS + gg];
        node = gg;
    }
    red[b] = -acc;
    __syncthreads();
#pragma unroll
    for (int s = BATCH / 2; s > 0; s >>= 1) {
        if (b < s) red[b] += red[b + s];
        __syncthreads();
    }
    if (b == 0) out[0] = red[0] * (1.0f / BATCH);
}

extern "C" void kernel_launch(void* const* d_in, const int* in_sizes, int n_in,
                              void* d_out, int out_size, void* d_ws, size_t ws_size,
                              hipStream_t stream) {
    (void)in_sizes; (void)n_in; (void)out_size; (void)ws_size;
    const float* scores      = (const float*)d_in[0];  // (1024, 9999) f32
    const int*   labels      = (const int*)d_in[1];    // (1024,)
    const int*   label_order = (const int*)d_in[5];    // (9375,) leaf ids
    float*       lse         = (float*)d_ws;           // (1024, 625) f32 = 2.56 MB

    dim3 blk(32, 8);                                   // 8 waves, one group each
    dim3 grd((NGROUPS + 7) / 8, BATCH / 16);
    lse_wmma_kernel<<<grd, blk, 0, stream>>>(scores, lse);
    path_nll_kernel<<<1, BATCH, 0, stream>>>(scores, labels, label_order, lse,
                                             (float*)d_out);
}